// SDEParallelFeatureHead_6975026888821
// MI455X (gfx1250) — compile-verified
//
#include <hip/hip_runtime.h>
#include <math.h>

// Problem constants (match reference)
#define BB 16
#define TT 128
#define HH 8
#define NN 64
#define OUT_DIM 64
#define FEAT 576            // N * 9
#define CHUNK 32            // t-chunk per block (stage 1)
#define NCHUNK (TT / CHUNK) // 4

static constexpr float S_EPS = 1e-8f;
// -EPS * ln(EPS) = 1e-8 * 18.420680743952367
static constexpr float CLIP_TERM = 1.8420680743952367e-7f;

typedef __attribute__((ext_vector_type(2))) float v2f;
typedef __attribute__((ext_vector_type(8))) float v8f;

// ---------------------------------------------------------------------------
// Stage 1: stream 256 MB of logits.
// Block = 256 threads = 8 waves (wave == head h).
// Each block owns (b, n-quad n0..n0+3, t-chunk). Each wave processes 4 rows
// per iteration: lanes [8r, 8r+8) hold row n0+r, 8 floats per lane
// (2 x global_load_b128 = 1 KB contiguous per wave per t).
// Reductions are 3-step butterflies within 8-lane groups.
// Produces ent(B,T,N), diff(B,T-1,N), diag(B,T,N), head-averaged in fixed
// order via LDS (deterministic, no float atomics).
// ---------------------------------------------------------------------------
__global__ __launch_bounds__(256) void stage1_stream(
    const float* __restrict__ logits,
    float* __restrict__ ent_out,
    float* __restrict__ diff_out,
    float* __restrict__ diag_out) {
  const int blk   = blockIdx.x;            // B * (N/4) * NCHUNK blocks
  const int chunk = blk & (NCHUNK - 1);
  const int bq    = blk >> 2;              // b*(N/4) + quad
  const int b     = bq >> 4;               // / (N/4 == 16)
  const int n0    = (bq & 15) * 4;         // base node row of this quad
  const int h     = threadIdx.x >> 5;      // wave index == head
  const int lane  = threadIdx.x & 31;
  const int r     = lane >> 3;             // row-in-quad handled by this lane
  const int j     = lane & 7;              // 8-float segment within the row

  __shared__ float s_ent[HH][4];
  __shared__ float s_dif[HH][4];
  __shared__ float s_dia[HH][4];

  const size_t tstride = (size_t)HH * NN * NN;   // elements between t
  // base of the 4-row group (b, t=0, h, n0, :)
  const float* base0 = logits + (((size_t)b * TT * HH + h) * NN + n0) * NN;
  const int fo = lane * 8;                 // this lane's float offset in 1KB

  const int t0     = chunk * CHUNK;
  const int tstart = (t0 == 0) ? 0 : t0 - 1;  // overlap row warms the diff chain
  const int tend   = t0 + CHUNK;

  float prev[8];
  #pragma unroll
  for (int i = 0; i < 8; ++i) prev[i] = 0.f;

  for (int t = tstart; t < tend; ++t) {
    const float* g = base0 + (size_t)t * tstride + fo;
    if (t + 1 < tend) {
      // global_prefetch_b8: next t-row group is 128 KB away
      __builtin_prefetch(g + tstride, 0, 1);
    }

    float x[8];
    *(float4*)(x)     = *(const float4*)(g);
    *(float4*)(x + 4) = *(const float4*)(g + 4);

    // --- row max (lane-local 8, then 3-step butterfly over 8-lane group) ---
    float m = x[0];
    #pragma unroll
    for (int i = 1; i < 8; ++i) m = fmaxf(m, x[i]);
    #pragma unroll
    for (int off = 4; off > 0; off >>= 1) m = fmaxf(m, __shfl_xor(m, off, 32));

    // --- exp and row sum ---
    float p[8];
    float s = 0.f;
    #pragma unroll
    for (int i = 0; i < 8; ++i) {
      x[i] -= m;               // keep shifted logit: q = lns - x[i] later
      p[i] = __expf(x[i]);
      s += p[i];
    }
    #pragma unroll
    for (int off = 4; off > 0; off >>= 1) s += __shfl_xor(s, off, 32);

    const float inv = 1.0f / s;
    const float lns = __logf(s);

    // --- entropy (with clip@1e-8 reproduction) and L1 temporal diff ---
    float ent = 0.f, d = 0.f;
    #pragma unroll
    for (int i = 0; i < 8; ++i) {
      p[i] *= inv;
      float q  = lns - x[i];           // == -ln(p_i)
      float pq = p[i] * q;
      pq = (p[i] < S_EPS) ? CLIP_TERM : pq;
      ent += pq;
      d += fabsf(p[i] - prev[i]);
      prev[i] = p[i];
    }
    #pragma unroll
    for (int off = 4; off > 0; off >>= 1) {
      ent += __shfl_xor(ent, off, 32);
      d   += __shfl_xor(d,   off, 32);
    }

    // --- self-loop prob: col (n0+r) of row (n0+r). Since n0 % 8 is 0 or 4,
    //     all 4 diag columns live in lane j == (n0>>3); comp = (n0&7) + r. ---
    float pdv;
    {
      float d0, d1, d2, d3;
      if ((n0 & 7) == 0) { d0 = p[0]; d1 = p[1]; d2 = p[2]; d3 = p[3]; }
      else               { d0 = p[4]; d1 = p[5]; d2 = p[6]; d3 = p[7]; }
      pdv = (r & 1) ? ((r & 2) ? d3 : d1) : ((r & 2) ? d2 : d0);
    }

    if (t >= t0) {                       // uniform branch (skip warm-up row)
      if (j == 0)         { s_ent[h][r] = ent; s_dif[h][r] = d; }
      if (j == (n0 >> 3)) { s_dia[h][r] = pdv; }
      __syncthreads();
      if (threadIdx.x < 4) {
        const int rr = threadIdx.x;
        float se = 0.f, sd = 0.f, sg = 0.f;   // fixed-order head sum
        #pragma unroll
        for (int i = 0; i < HH; ++i) {
          se += s_ent[i][rr]; sd += s_dif[i][rr]; sg += s_dia[i][rr];
        }
        const int oi = (b * TT + t) * NN + n0 + rr;
        ent_out[oi]  = se * (1.0f / HH);
        diag_out[oi] = sg * (1.0f / HH);
        if (t >= 1) diff_out[(b * (TT - 1) + (t - 1)) * NN + n0 + rr] = sd * (1.0f / HH);
      }
      __syncthreads();
    }
  }
}

// ---------------------------------------------------------------------------
// Stage 2: per-(b,n) temporal statistics -> feat (B, 576). grid=B, block=N.
// Reads are coalesced across n (stride-1 within a wave).
// ---------------------------------------------------------------------------
__global__ __launch_bounds__(NN) void stage2_stats(
    const float* __restrict__ ent,
    const float* __restrict__ diff,
    const float* __restrict__ diag,
    float* __restrict__ feat) {
  const int b = blockIdx.x;
  const int n = threadIdx.x;

  float es = 0.f, es2 = 0.f, emx = -INFINITY, emn = INFINITY, e0 = 0.f, eL = 0.f;
  for (int t = 0; t < TT; ++t) {
    float v = ent[(b * TT + t) * NN + n];
    es += v; es2 += v * v;
    emx = fmaxf(emx, v); emn = fminf(emn, v);
    if (t == 0) e0 = v;
    if (t == TT - 1) eL = v;
  }
  float emean = es * (1.0f / TT);
  float evar  = fmaxf(es2 * (1.0f / TT) - emean * emean, 0.0f);

  float ds = 0.f, ds2 = 0.f, dmx = -INFINITY;
  for (int t = 0; t < TT - 1; ++t) {
    float v = diff[(b * (TT - 1) + t) * NN + n];
    ds += v; ds2 += v * v; dmx = fmaxf(dmx, v);
  }
  float dmean = ds * (1.0f / (TT - 1));
  float dvar  = fmaxf(ds2 * (1.0f / (TT - 1)) - dmean * dmean, 0.0f);

  float gs = 0.f, gs2 = 0.f;
  for (int t = 0; t < TT; ++t) {
    float v = diag[(b * TT + t) * NN + n];
    gs += v; gs2 += v * v;
  }
  float gmean = gs * (1.0f / TT);
  float gvar  = fmaxf(gs2 * (1.0f / TT) - gmean * gmean, 0.0f);

  float* f = feat + b * FEAT + n * 9;
  f[0] = emean;
  f[1] = sqrtf(evar);
  f[2] = emx - emn;
  f[3] = (eL - e0) * (1.0f / (TT - 1));
  f[4] = dmean;
  f[5] = sqrtf(dvar);
  f[6] = dmx;
  f[7] = gmean;
  f[8] = sqrtf(gvar);
}

// ---------------------------------------------------------------------------
// Stage 3: LayerNorm(576) -> GEMM (16,576)x(576,64) via V_WMMA_F32_16X16X4_F32
// One block, 128 threads = 4 waves; wave w owns output N-tile [16w, 16w+16).
// ---------------------------------------------------------------------------
#define XW 577   // pad: 576 % 64 == 0 would cause 16-way LDS bank conflicts

__global__ __launch_bounds__(128) void stage3_ln_gemm(
    const float* __restrict__ feat,
    const float* __restrict__ gamma,
    const float* __restrict__ beta,
    const float* __restrict__ W,
    const float* __restrict__ bias,
    float* __restrict__ out) {
  __shared__ float X[BB][XW];
  __shared__ float s_mu[BB];
  __shared__ float s_rs[BB];

  const int tid = threadIdx.x;

  // load feat (16 x 576) into LDS
  for (int i = tid; i < BB * FEAT; i += 128) {
    int r = i / FEAT, c = i - r * FEAT;
    X[r][c] = feat[i];
  }
  __syncthreads();

  // per-row LayerNorm statistics
  if (tid < BB) {
    float s = 0.f, s2 = 0.f;
    for (int c = 0; c < FEAT; ++c) { float v = X[tid][c]; s += v; s2 += v * v; }
    float mu  = s * (1.0f / FEAT);
    float var = fmaxf(s2 * (1.0f / FEAT) - mu * mu, 0.0f);
    s_mu[tid] = mu;
    s_rs[tid] = rsqrtf(var + 1e-5f);
  }
  __syncthreads();

  // normalize + affine in LDS
  for (int i = tid; i < BB * FEAT; i += 128) {
    int r = i / FEAT, c = i - r * FEAT;
    X[r][c] = (X[r][c] - s_mu[r]) * s_rs[r] * gamma[c] + beta[c];
  }
  __syncthreads();

  // WMMA: A = X (16 x 576), B = W (576 x 64), tile K by 4.
  const int wave = tid >> 5;            // 0..3 -> N tile
  const int lane = tid & 31;
  const int n0   = wave * 16;
  const int lrow = lane & 15;           // M (A) / N (B) index within tile
  const int kh   = (lane >> 4) * 2;     // lanes 0-15 -> K=0,1 ; 16-31 -> K=2,3

  v8f acc = {};
  for (int k = 0; k < FEAT; k += 4) {
    v2f a, bb;
    a.x  = X[lrow][k + kh];
    a.y  = X[lrow][k + kh + 1];
    bb.x = W[(k + kh)     * OUT_DIM + n0 + lrow];
    bb.y = W[(k + kh + 1) * OUT_DIM + n0 + lrow];
    acc = __builtin_amdgcn_wmma_f32_16x16x4_f32(
        /*neg_a=*/false, a, /*neg_b=*/false, bb,
        /*c_mod=*/(short)0, acc, /*reuse_a=*/false, /*reuse_b=*/false);
  }

  // C/D layout: VGPR v -> (M = v + (lane>=16 ? 8 : 0), N = n0 + (lane & 15))
  const int cn = n0 + lrow;
  const float bv = bias[cn];
  #pragma unroll
  for (int v = 0; v < 8; ++v) {
    int row = v + ((lane >> 4) << 3);
    out[row * OUT_DIM + cn] = fmaxf(acc[v] + bv, 0.0f);
  }
}

// ---------------------------------------------------------------------------
extern "C" void kernel_launch(void* const* d_in, const int* in_sizes, int n_in,
                              void* d_out, int out_size, void* d_ws, size_t ws_size,
                              hipStream_t stream) {
  const float* logits = (const float*)d_in[0]; // (B,T,H,N,N)
  const float* gamma  = (const float*)d_in[1]; // (576)
  const float* beta   = (const float*)d_in[2]; // (576)
  const float* W      = (const float*)d_in[3]; // (576,64)
  const float* bias   = (const float*)d_in[4]; // (64)
  float* out = (float*)d_out;                  // (16,64)

  float* ws   = (float*)d_ws;
  float* ent  = ws;                                   // B*T*N
  float* diff = ent  + (size_t)BB * TT * NN;          // B*(T-1)*N
  float* diag = diff + (size_t)BB * (TT - 1) * NN;    // B*T*N
  float* feat = diag + (size_t)BB * TT * NN;          // B*576
  // total: ~1.6 MB of d_ws; every element is written each call (no init needed)

  stage1_stream<<<dim3(BB * (NN / 4) * NCHUNK), dim3(256), 0, stream>>>(
      logits, ent, diff, diag);
  stage2_stats <<<dim3(BB), dim3(NN),  0, stream>>>(ent, diff, diag, feat);
  stage3_ln_gemm<<<dim3(1), dim3(128), 0, stream>>>(feat, gamma, beta, W, bias, out);
}